// GraphConv_11269994185513
// MI455X (gfx1250) — compile-verified
//
#include <hip/hip_runtime.h>
#include <hip/hip_bf16.h>

// ---------------------------------------------------------------------------
// GraphConv on MI455X (gfx1250, wave32):
//   h   = x @ w                    (WMMA f32 16x16x4, full f32 precision)
//   out = relu(scatter_add(edge_weight * h[edge_src] -> edge_dst))
//
// Problem constants from the reference.
// ---------------------------------------------------------------------------
#define N_NODES 50000
#define N_EDGES 800000
#define IN_DIM  128
#define OUT_DIM 64

typedef __attribute__((ext_vector_type(2))) float v2f;
typedef __attribute__((ext_vector_type(8))) float v8f;

// ---------------------------------------------------------------------------
// Kernel 1: h = x @ w via V_WMMA_F32_16X16X4_F32.
//   block = 128 threads = 4 waves; wave w -> N-tile w (N = 4*16 = 64)
//   blockIdx.x -> M-tile (50000 / 16 = 3125 tiles, exact)
//   K = 128 = 32 steps of 4.
// A-matrix layout (32-bit A 16x4): lane l<16 holds {K=k, K=k+1} for row M=l,
// lanes 16..31 hold {K=k+2, K=k+3} -> one global_load_b64 per lane per step.
// B-matrix mirrors with N instead of M. C/D: VGPR r = rows {r, 8+r}.
// EXEC is all-ones everywhere (no divergence) as WMMA requires.
// ---------------------------------------------------------------------------
__global__ __launch_bounds__(128)
void gcn_gemm_wmma(const float* __restrict__ x,
                   const float* __restrict__ w,
                   float* __restrict__ h) {
    const int lane = threadIdx.x & 31;
    const int wave = threadIdx.x >> 5;       // 0..3 -> N-tile
    const int half = lane >> 4;              // 0 or 1 (K-split within wave)
    const int l    = lane & 15;              // row (A) / col (B) within tile

    const int m0 = blockIdx.x << 4;          // 16-row M tile
    const int n0 = wave << 4;                // 16-col N tile

    const float* __restrict__ arow = x + (size_t)(m0 + l) * IN_DIM;

    v8f c = {};
    #pragma unroll 4
    for (int k = 0; k < IN_DIM; k += 4) {
        // A: two consecutive K values for this lane's row -> 8B load
        v2f a = *(const v2f*)(arow + k + 2 * half);
        // B: w[k + 2*half + {0,1}][n0 + l]
        v2f b;
        b.x = w[(size_t)(k + 2 * half)     * OUT_DIM + n0 + l];
        b.y = w[(size_t)(k + 2 * half + 1) * OUT_DIM + n0 + l];
        // 8-arg form: (neg_a, A, neg_b, B, c_mod, C, reuse_a, reuse_b)
        c = __builtin_amdgcn_wmma_f32_16x16x4_f32(
                false, a, false, b, (short)0, c, false, false);
    }

    // D layout: VGPR r -> row m0 + 8*half + r, col n0 + l
    float* __restrict__ hout = h + (size_t)(m0 + 8 * half) * OUT_DIM + n0 + l;
    #pragma unroll
    for (int r = 0; r < 8; ++r)
        hout[(size_t)r * OUT_DIM] = c[r];
}

// ---------------------------------------------------------------------------
// Kernel 2: zero the accumulator (d_out). float4 stores, exact coverage:
// 50000*64 = 3,200,000 floats = 800,000 float4.
// ---------------------------------------------------------------------------
__global__ __launch_bounds__(256)
void gcn_zero(float4* __restrict__ out, int n4) {
    int i = blockIdx.x * blockDim.x + threadIdx.x;
    if (i < n4) out[i] = make_float4(0.f, 0.f, 0.f, 0.f);
}

// ---------------------------------------------------------------------------
// Kernel 3: edge scatter. 32 threads per edge, 2 columns per thread (float2).
// out[dst] += weight * h[src]; f32 atomics resolve in L2 (h and out are both
// L2-resident at 12.8 MB each on a 192 MB L2). unsafeAtomicAdd lowers to
// global_atomic_add_f32 (no CAS loop).
// ---------------------------------------------------------------------------
__global__ __launch_bounds__(256)
void gcn_scatter(const float* __restrict__ h,
                 const int*   __restrict__ esrc,
                 const int*   __restrict__ edst,
                 const float* __restrict__ ew,
                 float* __restrict__ out) {
    unsigned gid = blockIdx.x * blockDim.x + threadIdx.x;
    unsigned e   = gid >> 5;               // edge id
    if (e >= N_EDGES) return;
    int cc = (gid & 31) * 2;               // column pair

    int   s  = esrc[e];
    int   d  = edst[e];
    float wt = ew[e];

    const float2 v = *(const float2*)(h + (size_t)s * OUT_DIM + cc);
    float* o = out + (size_t)d * OUT_DIM + cc;
    unsafeAtomicAdd(o,     v.x * wt);
    unsafeAtomicAdd(o + 1, v.y * wt);
}

// ---------------------------------------------------------------------------
// Kernel 4: in-place ReLU on d_out.
// ---------------------------------------------------------------------------
__global__ __launch_bounds__(256)
void gcn_relu(float4* __restrict__ out, int n4) {
    int i = blockIdx.x * blockDim.x + threadIdx.x;
    if (i < n4) {
        float4 v = out[i];
        v.x = fmaxf(v.x, 0.f);
        v.y = fmaxf(v.y, 0.f);
        v.z = fmaxf(v.z, 0.f);
        v.w = fmaxf(v.w, 0.f);
        out[i] = v;
    }
}

// ---------------------------------------------------------------------------
// Host-side launcher. Input order (setup_inputs): x, edge_src, edge_dst,
// edge_weight, w. Output: [N_NODES, OUT_DIM] f32. d_ws holds h (12.8 MB).
// ---------------------------------------------------------------------------
extern "C" void kernel_launch(void* const* d_in, const int* in_sizes, int n_in,
                              void* d_out, int out_size, void* d_ws, size_t ws_size,
                              hipStream_t stream) {
    const float* x    = (const float*)d_in[0];
    const int*   esrc = (const int*)  d_in[1];
    const int*   edst = (const int*)  d_in[2];
    const float* ew   = (const float*)d_in[3];
    const float* w    = (const float*)d_in[4];
    float*       out  = (float*)d_out;
    float*       h    = (float*)d_ws;      // [N_NODES, OUT_DIM] scratch

    (void)in_sizes; (void)n_in; (void)out_size; (void)ws_size;

    // 1) h = x @ w   (3125 M-tiles x 4 N-tile waves)
    gcn_gemm_wmma<<<N_NODES / 16, 128, 0, stream>>>(x, w, h);

    // 2) out = 0
    const int n4 = (N_NODES * OUT_DIM) / 4;          // 800,000
    gcn_zero<<<(n4 + 255) / 256, 256, 0, stream>>>((float4*)out, n4);

    // 3) out[dst] += weight * h[src]
    const unsigned scatter_threads = (unsigned)N_EDGES * 32u;  // 25.6M
    gcn_scatter<<<scatter_threads / 256, 256, 0, stream>>>(h, esrc, edst, ew, out);

    // 4) out = relu(out)
    gcn_relu<<<(n4 + 255) / 256, 256, 0, stream>>>((float4*)out, n4);
}